// GraphNet_25168508354593
// MI455X (gfx1250) — compile-verified
//
#include <hip/hip_runtime.h>
#include <hip/hip_bf16.h>

typedef float v2f __attribute__((ext_vector_type(2)));
typedef float v8f __attribute__((ext_vector_type(8)));

#define WMMA_F32X4(C, A, B) \
  (C) = __builtin_amdgcn_wmma_f32_16x16x4_f32(false, (A), false, (B), (short)0, (C), false, false)

// ---------------------------------------------------------------------------
// k_repack: rearrange W[K x C] into the WMMA B-operand lane layout so each
// lane reads its (k0, k0+1) pair with ONE coalesced global_load_b64.
//   pair index = ((s * (C/16) + t) * 32 + lane), s = k/4, t = column tile
//   lane: n = (lane&15) + 16*t, kb = 2*(lane>>4), k0 = 4*s + kb
// ---------------------------------------------------------------------------
__global__ void k_repack(const float* __restrict__ W, float* __restrict__ Wp,
                         int K, int C)
{
    int idx   = blockIdx.x * blockDim.x + threadIdx.x;
    int nT    = C >> 4;
    int total = (K >> 2) * nT * 32;
    if (idx >= total) return;
    int lane = idx & 31;
    int rest = idx >> 5;
    int t    = rest % nT;
    int s    = rest / nT;
    int n    = (lane & 15) + (t << 4);
    int k0   = (s << 2) + ((lane >> 4) << 1);
    Wp[idx * 2 + 0] = W[(size_t)k0 * C + n];
    Wp[idx * 2 + 1] = W[(size_t)(k0 + 1) * C + n];
}

// ---------------------------------------------------------------------------
// k_proj<K, FUSED_BN>: Z[N x 32] = op(A[N x K]) @ W[K x 32]
//   FUSED_BN folds the layer-1 BatchNorm (x*scale[c]+shift[c]) into the load.
// One wave computes a 16-row x 32-col tile with two f32 WMMA accumulators.
// Requires N % 16 == 0 (true: N = 100000).
// ---------------------------------------------------------------------------
template <int K, bool FUSED_BN>
__global__ void k_proj(const float* __restrict__ A, const float* __restrict__ Wp,
                       const float* __restrict__ scale, const float* __restrict__ shift,
                       float* __restrict__ Z, int nNodes)
{
    int wave = blockIdx.x * (blockDim.x >> 5) + (threadIdx.x >> 5);
    int lane = threadIdx.x & 31;
    int row0 = wave << 4;
    if (row0 >= nNodes) return;                 // wave-uniform exit

    int m  = lane & 15;
    int hi = lane >> 4;
    int kb = hi << 1;
    const float* Arow = A + (size_t)(row0 + m) * K;
    const v2f*   Wp2  = (const v2f*)Wp;

    v8f c0 = {0.f,0.f,0.f,0.f,0.f,0.f,0.f,0.f};
    v8f c1 = {0.f,0.f,0.f,0.f,0.f,0.f,0.f,0.f};

    for (int k = 0; k < K; k += 4) {
        int k0 = k + kb;
        v2f a;
        if (FUSED_BN) {
            a.x = Arow[k0]     * scale[k0]     + shift[k0];
            a.y = Arow[k0 + 1] * scale[k0 + 1] + shift[k0 + 1];
        } else {
            a.x = Arow[k0];
            a.y = Arow[k0 + 1];
        }
        int s  = k >> 2;
        v2f b0 = Wp2[(s * 2 + 0) * 32 + lane];
        v2f b1 = Wp2[(s * 2 + 1) * 32 + lane];
        WMMA_F32X4(c0, a, b0);
        WMMA_F32X4(c1, a, b1);
    }

#pragma unroll
    for (int r = 0; r < 8; ++r) {
        int rr = row0 + r + (hi << 3);
        Z[(size_t)rr * 32 + m]      = c0[r];
        Z[(size_t)rr * 32 + 16 + m] = c1[r];
    }
}

// ---------------------------------------------------------------------------
// k_scatter: agg[dst[e]][0..31] += Z[src[e]][0..31], one wave per edge.
// Z and agg are 12.8 MB each -> L2-resident gathers and L2 f32 atomics.
// edge_index layout (2, E) row-major: src = ei[e], dst = ei[E + e].
// ---------------------------------------------------------------------------
__global__ void k_scatter(const float* __restrict__ Z, const int* __restrict__ ei,
                          float* __restrict__ agg, int E)
{
    int e    = blockIdx.x * (blockDim.x >> 5) + (threadIdx.x >> 5);
    int lane = threadIdx.x & 31;
    if (e >= E) return;
    int s = ei[e];
    int d = ei[E + e];
    atomicAdd(&agg[(size_t)d * 32 + lane], Z[(size_t)s * 32 + lane]);
}

// ---------------------------------------------------------------------------
// k_mlp1: per node  u = relu(Z + agg + b1a)  (A-matrix built on the fly),
//         H = relu(u @ W1b + b1b), plus per-column sum/sumsq atomics for BN.
// ---------------------------------------------------------------------------
__global__ void k_mlp1(const float* __restrict__ Z, const float* __restrict__ agg,
                       const float* __restrict__ b1a, const float* __restrict__ Wp,
                       const float* __restrict__ b1b, float* __restrict__ H,
                       float* __restrict__ s_sum, float* __restrict__ s_sq, int nNodes)
{
    int wave = blockIdx.x * (blockDim.x >> 5) + (threadIdx.x >> 5);
    int lane = threadIdx.x & 31;
    int row0 = wave << 4;
    if (row0 >= nNodes) return;

    int m  = lane & 15;
    int hi = lane >> 4;
    int kb = hi << 1;
    size_t rbase = (size_t)(row0 + m) * 32;
    const v2f* Wp2 = (const v2f*)Wp;

    v8f c0 = {0.f,0.f,0.f,0.f,0.f,0.f,0.f,0.f};
    v8f c1 = {0.f,0.f,0.f,0.f,0.f,0.f,0.f,0.f};

#pragma unroll
    for (int k = 0; k < 32; k += 4) {
        int k0 = k + kb;
        v2f a;
        a.x = fmaxf(Z[rbase + k0]     + agg[rbase + k0]     + b1a[k0],     0.f);
        a.y = fmaxf(Z[rbase + k0 + 1] + agg[rbase + k0 + 1] + b1a[k0 + 1], 0.f);
        int s  = k >> 2;
        v2f b0 = Wp2[(s * 2 + 0) * 32 + lane];
        v2f b1 = Wp2[(s * 2 + 1) * 32 + lane];
        WMMA_F32X4(c0, a, b0);
        WMMA_F32X4(c1, a, b1);
    }

    float bb0 = b1b[m], bb1 = b1b[16 + m];
    float s0 = 0.f, q0 = 0.f, s1 = 0.f, q1 = 0.f;
#pragma unroll
    for (int r = 0; r < 8; ++r) {
        int rr = row0 + r + (hi << 3);
        float v0 = fmaxf(c0[r] + bb0, 0.f);
        float v1 = fmaxf(c1[r] + bb1, 0.f);
        H[(size_t)rr * 32 + m]      = v0;
        H[(size_t)rr * 32 + 16 + m] = v1;
        s0 += v0; q0 += v0 * v0;
        s1 += v1; q1 += v1 * v1;
    }
    // lane L and L+16 hold the same output column -> fold halves, then atomics
    s0 += __shfl_xor(s0, 16, 32);  q0 += __shfl_xor(q0, 16, 32);
    s1 += __shfl_xor(s1, 16, 32);  q1 += __shfl_xor(q1, 16, 32);
    if (hi == 0) {
        atomicAdd(&s_sum[m], s0);       atomicAdd(&s_sq[m], q0);
        atomicAdd(&s_sum[16 + m], s1);  atomicAdd(&s_sq[16 + m], q1);
    }
}

// ---------------------------------------------------------------------------
// k_mlp2: per node  u = relu(Z + agg + b2a);  out = relu(u @ W2b + b2b)
// One wave: 16 rows x 128 cols (8 WMMA accumulators). Writes pre-BN result
// straight into d_out and accumulates BN stats for the 128 columns.
// ---------------------------------------------------------------------------
__global__ void k_mlp2(const float* __restrict__ Z, const float* __restrict__ agg,
                       const float* __restrict__ b2a, const float* __restrict__ Wp,
                       const float* __restrict__ b2b, float* __restrict__ out,
                       float* __restrict__ s_sum, float* __restrict__ s_sq, int nNodes)
{
    int wave = blockIdx.x * (blockDim.x >> 5) + (threadIdx.x >> 5);
    int lane = threadIdx.x & 31;
    int row0 = wave << 4;
    if (row0 >= nNodes) return;

    int m  = lane & 15;
    int hi = lane >> 4;
    int kb = hi << 1;
    size_t rbase = (size_t)(row0 + m) * 32;
    const v2f* Wp2 = (const v2f*)Wp;

    v8f c[8];
#pragma unroll
    for (int t = 0; t < 8; ++t) c[t] = (v8f){0.f,0.f,0.f,0.f,0.f,0.f,0.f,0.f};

#pragma unroll
    for (int k = 0; k < 32; k += 4) {
        int k0 = k + kb;
        v2f a;
        a.x = fmaxf(Z[rbase + k0]     + agg[rbase + k0]     + b2a[k0],     0.f);
        a.y = fmaxf(Z[rbase + k0 + 1] + agg[rbase + k0 + 1] + b2a[k0 + 1], 0.f);
        int s = k >> 2;
#pragma unroll
        for (int t = 0; t < 8; ++t) {
            v2f b = Wp2[(s * 8 + t) * 32 + lane];
            WMMA_F32X4(c[t], a, b);
        }
    }

#pragma unroll
    for (int t = 0; t < 8; ++t) {
        float bb = b2b[t * 16 + m];
        float s = 0.f, q = 0.f;
#pragma unroll
        for (int r = 0; r < 8; ++r) {
            int rr = row0 + r + (hi << 3);
            float v = fmaxf(c[t][r] + bb, 0.f);
            out[(size_t)rr * 128 + t * 16 + m] = v;
            s += v; q += v * v;
        }
        s += __shfl_xor(s, 16, 32);
        q += __shfl_xor(q, 16, 32);
        if (hi == 0) {
            atomicAdd(&s_sum[t * 16 + m], s);
            atomicAdd(&s_sq[t * 16 + m], q);
        }
    }
}

// Turn (sum, sumsq) into fused BN scale/shift: y = x*scale + shift
__global__ void k_finalize(const float* __restrict__ sum, const float* __restrict__ sq,
                           const float* __restrict__ g, const float* __restrict__ be,
                           float* __restrict__ scale, float* __restrict__ shift,
                           int C, float invN)
{
    int cidx = blockIdx.x * blockDim.x + threadIdx.x;
    if (cidx >= C) return;
    float mean = sum[cidx] * invN;
    float var  = fmaxf(sq[cidx] * invN - mean * mean, 0.f);   // biased variance (torch-style)
    float s    = g[cidx] * rsqrtf(var + 1e-5f);
    scale[cidx] = s;
    shift[cidx] = be[cidx] - mean * s;
}

// In-place BN epilogue on d_out (N x 128)
__global__ void k_bnapply(float* __restrict__ out, const float* __restrict__ scale,
                          const float* __restrict__ shift, size_t total)
{
    size_t i = (size_t)blockIdx.x * blockDim.x + threadIdx.x;
    if (i >= total) return;
    int c = (int)(i & 127);
    out[i] = out[i] * scale[c] + shift[c];
}

// ---------------------------------------------------------------------------
extern "C" void kernel_launch(void* const* d_in, const int* in_sizes, int n_in,
                              void* d_out, int out_size, void* d_ws, size_t ws_size,
                              hipStream_t stream)
{
    const float* x   = (const float*)d_in[0];
    const int*   ei  = (const int*)d_in[1];     // edge_index (2, E)
    const float* W1a = (const float*)d_in[2];
    const float* b1a = (const float*)d_in[3];
    const float* W1b = (const float*)d_in[4];
    const float* b1b = (const float*)d_in[5];
    const float* g1  = (const float*)d_in[6];
    const float* be1 = (const float*)d_in[7];
    const float* W2a = (const float*)d_in[8];
    const float* b2a = (const float*)d_in[9];
    const float* W2b = (const float*)d_in[10];
    const float* b2b = (const float*)d_in[11];
    const float* g2  = (const float*)d_in[12];
    const float* be2 = (const float*)d_in[13];

    const int N = in_sizes[0] / 128;            // 100000
    const int E = in_sizes[1] / 2;              // 1600000
    float* out = (float*)d_out;

    // Workspace layout (floats):
    //   z(N*32) | agg(N*32) | h1(N*32) | stats(640) | repacked weights(10240)
    float* ws   = (float*)d_ws;
    float* z    = ws;
    float* agg  = ws + (size_t)N * 32;
    float* h1   = ws + (size_t)N * 64;
    float* st   = ws + (size_t)N * 96;
    float* sum1 = st;         float* sq1 = st + 32;
    float* sum2 = st + 64;    float* sq2 = st + 192;
    float* sc1  = st + 320;   float* sh1 = st + 352;
    float* sc2  = st + 384;   float* sh2 = st + 512;
    float* wp1a = st + 640;                    // (128/4)*2*32*2 = 4096
    float* wp1b = wp1a + 4096;                 // (32/4)*2*32*2  = 1024
    float* wp2a = wp1b + 1024;                 // (32/4)*2*32*2  = 1024
    float* wp2b = wp2a + 1024;                 // (32/4)*8*32*2  = 4096

    const dim3 blk(256);                        // 8 wave32s / block
    const int rowWaves = (N + 15) / 16;         // 6250
    const int gemmGrid = (rowWaves + 7) / 8;
    const int scatGrid = (E + 7) / 8;           // one wave per edge
    const float invN   = 1.0f / (float)N;

    // ---- weight repack (tiny, once per launch) -----------------------------
    k_repack<<<8, blk, 0, stream>>>(W1a, wp1a, 128, 32);
    k_repack<<<2, blk, 0, stream>>>(W1b, wp1b, 32, 32);
    k_repack<<<2, blk, 0, stream>>>(W2a, wp2a, 32, 32);
    k_repack<<<8, blk, 0, stream>>>(W2b, wp2b, 32, 128);

    // ---- layer 1 -----------------------------------------------------------
    hipMemsetAsync(agg, 0, (size_t)N * 32 * sizeof(float), stream);
    hipMemsetAsync(st,  0, 320 * sizeof(float), stream);

    // z = x @ W1a   (project BEFORE aggregating: 4x less scatter traffic,
    //                and the 12.8MB feature table stays L2-resident)
    k_proj<128, false><<<gemmGrid, blk, 0, stream>>>(x, wp1a, nullptr, nullptr, z, N);
    // agg = scatter_add(z[src] -> dst)
    k_scatter<<<scatGrid, blk, 0, stream>>>(z, ei, agg, E);
    // h1 = relu(relu(z + agg + b1a) @ W1b + b1b), accumulate BN1 stats
    k_mlp1<<<gemmGrid, blk, 0, stream>>>(z, agg, b1a, wp1b, b1b, h1, sum1, sq1, N);
    k_finalize<<<1, 32, 0, stream>>>(sum1, sq1, g1, be1, sc1, sh1, 32, invN);

    // ---- layer 2 -----------------------------------------------------------
    hipMemsetAsync(agg, 0, (size_t)N * 32 * sizeof(float), stream);

    // z = BN1(h1) @ W2a   (BN fused into the projection; z buffer reused)
    k_proj<32, true><<<gemmGrid, blk, 0, stream>>>(h1, wp2a, sc1, sh1, z, N);
    k_scatter<<<scatGrid, blk, 0, stream>>>(z, ei, agg, E);
    // out = relu(relu(z + agg + b2a) @ W2b + b2b) pre-BN, accumulate BN2 stats
    k_mlp2<<<gemmGrid, blk, 0, stream>>>(z, agg, b2a, wp2b, b2b, out, sum2, sq2, N);
    k_finalize<<<1, 128, 0, stream>>>(sum2, sq2, g2, be2, sc2, sh2, 128, invN);

    // in-place BN epilogue on output
    const size_t total = (size_t)N * 128;
    k_bnapply<<<(int)((total + 255) / 256), blk, 0, stream>>>(out, sc2, sh2, total);
}